// Model_23965917512140
// MI455X (gfx1250) — compile-verified
//
#include <hip/hip_runtime.h>
#include <hip/hip_bf16.h>
#include <math.h>

typedef __attribute__((ext_vector_type(2))) float v2f;
typedef __attribute__((ext_vector_type(8))) float v8f;

// Problem dims (fixed by the reference setup)
constexpr int Bx = 2, Tt = 32, Hh = 16, Ww = 16, Dd = 128, Ee = 9, Ff = 16;
constexpr int Nn = Bx * Hh * Ww;          // 512 tokens
constexpr int Mrows = Nn * Tt;            // 16384 rows of xp
constexpr int LDSP = 132;                 // padded row (floats): conflict-free col reads, 528B (16B-aligned pitch)

#define EPSV 2.2204460492503131e-16f      // np.finfo(float64).eps

// ---------------------------------------------------------------------------
// CDNA5 async global->LDS copy (ASYNCcnt-tracked, no VGPR round trip).
// LDS byte address = low 32 bits of the flat pointer (LDS aperture: addr[31:0]).
// ---------------------------------------------------------------------------
__device__ __forceinline__ void async_copy_b128(uint32_t lds_addr, const void* gptr) {
  asm volatile("global_load_async_to_lds_b128 %0, %1, off"
               :
               : "v"(lds_addr), "v"((unsigned long long)(uintptr_t)gptr)
               : "memory");
}

__device__ __forceinline__ void wait_async0() {
#if __has_builtin(__builtin_amdgcn_s_wait_asynccnt)
  __builtin_amdgcn_s_wait_asynccnt(0);
#else
  asm volatile("s_wait_asynccnt 0x0" ::: "memory");
#endif
}

// ---------------------------------------------------------------------------
// Kernel 1: xp[n*T+t, d] = sum_k x[b,t,0,h,w,k] * w_fuse[k,d] + b_fuse[d]
// 1024 blocks (16-row tiles), 256 threads = 8 waves, wave w owns col-tile w.
// ---------------------------------------------------------------------------
__global__ __launch_bounds__(256) void k_fuse_proj(const float* __restrict__ x,
                                                   const float* __restrict__ w_fuse,
                                                   const float* __restrict__ b_fuse,
                                                   float* __restrict__ xp) {
  __shared__ float As[16 * LDSP];
  const int rt  = blockIdx.x;        // row tile
  const int tid = threadIdx.x;

  // async stage of 16x128 A tile (gather from x with the b,h,w,t transpose)
  {
    const int m  = tid >> 4;         // 0..15 row in tile
    const int kc = (tid & 15) * 8;   // 8 floats (32B) per thread
    const int row = rt * 16 + m;
    const int n = row >> 5;          // row / T
    const int t = row & 31;
    const int b = n >> 8;            // n / (H*W)
    const int h = (n >> 4) & 15;
    const int w = n & 15;
    const float* src = x + ((((size_t)(b * Tt + t) * Hh + h) * Ww + w) * Dd + kc);
    const uint32_t l0 = (uint32_t)(uintptr_t)(&As[m * LDSP + kc]);
    async_copy_b128(l0,      src);
    async_copy_b128(l0 + 16, src + 4);
  }
  wait_async0();
  __syncthreads();

  const int wv = tid >> 5;           // wave id = col tile 0..7
  const int l  = tid & 31;
  const int lm = l & 15;
  const int lh = l >> 4;

  v8f c = {};
  const float* Bb = w_fuse + wv * 16 + lm;         // column lm of this col-tile
  const float* Ar = &As[lm * LDSP];

  for (int k0 = 0; k0 < Dd; k0 += 4) {
    const int koff = k0 + 2 * lh;
    v2f a, bfr;
    a.x   = Ar[koff];
    a.y   = Ar[koff + 1];
    bfr.x = Bb[(size_t)koff * Dd];
    bfr.y = Bb[(size_t)(koff + 1) * Dd];
    c = __builtin_amdgcn_wmma_f32_16x16x4_f32(false, a, false, bfr,
                                              (short)0, c, false, false);
  }

  const int col  = wv * 16 + lm;
  const float bias = b_fuse[col];
#pragma unroll
  for (int v = 0; v < 8; ++v) {
    const int m = v + 8 * lh;
    xp[(size_t)(rt * 16 + m) * Dd + col] = c[v] + bias;
  }
}

// ---------------------------------------------------------------------------
// Kernel 2: gating. 512 blocks (one per n), 128 threads (one per d).
// 32-pt real DFT (freqs 1..16, ortho), amp @ w_gate, mean over d,
// top-2 (stable, desc) + softmax -> (g0,g1,i0,i1).
// ---------------------------------------------------------------------------
__global__ __launch_bounds__(128) void k_gate(const float* __restrict__ xp,
                                              const float* __restrict__ w_gate,
                                              float* __restrict__ gate_g,
                                              int* __restrict__ gate_i) {
  __shared__ float red[Ee][128];
  __shared__ float wts[Ee];
  const int n = blockIdx.x;
  const int d = threadIdx.x;

  float xs[Tt];
  const float* base = xp + (size_t)n * Tt * Dd + d;
#pragma unroll
  for (int t = 0; t < Tt; ++t) xs[t] = base[t * Dd];

  float lacc[Ee];
#pragma unroll
  for (int e = 0; e < Ee; ++e) lacc[e] = 0.f;

  const float inv_sqrtT = 0.17677669529663688f;  // 1/sqrt(32), ortho norm
  for (int f = 1; f <= Ff; ++f) {
    float re = 0.f, im = 0.f;
    for (int t = 0; t < Tt; ++t) {
      const float ang = -6.283185307179586f * (float)(f * t) * (1.0f / 32.0f);
      float s, cth;
      __sincosf(ang, &s, &cth);
      re += xs[t] * cth;
      im += xs[t] * s;
    }
    const float amp = sqrtf(re * re + im * im) * inv_sqrtT;
#pragma unroll
    for (int e = 0; e < Ee; ++e) lacc[e] += amp * w_gate[(f - 1) * Ee + e];
  }

#pragma unroll
  for (int e = 0; e < Ee; ++e) red[e][d] = lacc[e];
  __syncthreads();

  if (d < Ee) {
    float s = 0.f;
    for (int i = 0; i < 128; ++i) s += red[d][i];
    wts[d] = s * (1.0f / 128.0f);    // mean over d
  }
  __syncthreads();

  if (d == 0) {
    int i0 = 0, i1 = 0;
    float w0 = -INFINITY, w1 = -INFINITY;
    for (int e = 0; e < Ee; ++e) {
      const float v = wts[e];
      if (v > w0)      { w1 = w0; i1 = i0; w0 = v; i0 = e; }
      else if (v > w1) { w1 = v; i1 = e; }
    }
    float g0 = expf(w0 - w0);
    float g1 = expf(w1 - w0);
    const float s = g0 + g1;
    g0 /= s; g1 /= s;
    gate_g[n * 2 + 0] = g0;  gate_g[n * 2 + 1] = g1;
    gate_i[n * 2 + 0] = i0;  gate_i[n * 2 + 1] = i1;
  }
}

// ---------------------------------------------------------------------------
// Kernel 3: routed experts + combine.
// 512 blocks (one per n), 512 threads = 16 waves = 2 row-tiles x 8 col-tiles.
// Each wave accumulates BOTH routed experts (two v8f accumulators), then
// out = log(g0*exp(y0) + g1*exp(y1)) with the eps floor. Single store.
// ---------------------------------------------------------------------------
__global__ __launch_bounds__(512) void k_experts(const float* __restrict__ xp,
                                                 const float* __restrict__ We,
                                                 const float* __restrict__ be,
                                                 const float* __restrict__ gate_g,
                                                 const int* __restrict__ gate_i,
                                                 float* __restrict__ out) {
  __shared__ float As[32 * LDSP];
  const int n   = blockIdx.x;
  const int tid = threadIdx.x;

  // async stage of xp[n] : 32x128 into LDS (padded rows)
  {
    const int m  = tid >> 4;         // 0..31
    const int kc = (tid & 15) * 8;
    const float* src = xp + (size_t)(n * Tt + m) * Dd + kc;
    const uint32_t l0 = (uint32_t)(uintptr_t)(&As[m * LDSP + kc]);
    async_copy_b128(l0,      src);
    async_copy_b128(l0 + 16, src + 4);
  }
  wait_async0();
  __syncthreads();

  const int e0 = gate_i[n * 2 + 0];
  const int e1 = gate_i[n * 2 + 1];
  const float g0 = gate_g[n * 2 + 0];
  const float g1 = gate_g[n * 2 + 1];

  const int wv = tid >> 5;           // 0..15
  const int rt = wv >> 3;            // row tile 0..1
  const int ct = wv & 7;             // col tile 0..7
  const int l  = tid & 31;
  const int lm = l & 15;
  const int lh = l >> 4;

  v8f c0 = {}, c1 = {};
  const float* B0 = We + (size_t)e0 * Dd * Dd + ct * 16 + lm;
  const float* B1 = We + (size_t)e1 * Dd * Dd + ct * 16 + lm;
  const float* Ar = &As[(rt * 16 + lm) * LDSP];

  for (int k0 = 0; k0 < Dd; k0 += 4) {
    const int koff = k0 + 2 * lh;
    v2f a, b0, b1;
    a.x  = Ar[koff];
    a.y  = Ar[koff + 1];
    b0.x = B0[(size_t)koff * Dd];
    b0.y = B0[(size_t)(koff + 1) * Dd];
    b1.x = B1[(size_t)koff * Dd];
    b1.y = B1[(size_t)(koff + 1) * Dd];
    c0 = __builtin_amdgcn_wmma_f32_16x16x4_f32(false, a, false, b0,
                                               (short)0, c0, false, false);
    c1 = __builtin_amdgcn_wmma_f32_16x16x4_f32(false, a, false, b1,
                                               (short)0, c1, false, false);
  }

  const int col = ct * 16 + lm;
  const float bb0 = be[e0 * Dd + col];
  const float bb1 = be[e1 * Dd + col];
#pragma unroll
  for (int v = 0; v < 8; ++v) {
    const int m = v + 8 * lh;
    const int t = rt * 16 + m;
    const float y0 = c0[v] + bb0;
    const float y1 = c1[v] + bb1;
    float comb = g0 * expf(y0) + g1 * expf(y1);
    if (comb == 0.f) comb = EPSV;
    out[((size_t)n * Tt + t) * Dd + col] = logf(comb);
  }
}

// ---------------------------------------------------------------------------
extern "C" void kernel_launch(void* const* d_in, const int* in_sizes, int n_in,
                              void* d_out, int out_size, void* d_ws, size_t ws_size,
                              hipStream_t stream) {
  (void)in_sizes; (void)n_in; (void)out_size; (void)ws_size;
  const float* x      = (const float*)d_in[0];
  const float* w_fuse = (const float*)d_in[1];
  const float* b_fuse = (const float*)d_in[2];
  const float* w_gate = (const float*)d_in[3];
  const float* We     = (const float*)d_in[4];
  const float* be     = (const float*)d_in[5];
  float* out = (float*)d_out;

  // workspace layout: xp (8 MB) | gate_g (4 KB) | gate_i (4 KB)
  float* xp     = (float*)d_ws;
  float* gate_g = xp + (size_t)Mrows * Dd;
  int*   gate_i = (int*)(gate_g + Nn * 2);

  k_fuse_proj<<<Mrows / 16, 256, 0, stream>>>(x, w_fuse, b_fuse, xp);
  k_gate<<<Nn, 128, 0, stream>>>(xp, w_gate, gate_g, gate_i);
  k_experts<<<Nn, 512, 0, stream>>>(xp, We, be, gate_g, gate_i, out);
}